// NL_79542794322615
// MI455X (gfx1250) — compile-verified
//
#include <hip/hip_runtime.h>
#include <hip/hip_bf16.h>

typedef float v2f __attribute__((ext_vector_type(2)));
typedef float v8f __attribute__((ext_vector_type(8)));
typedef int   i32x4 __attribute__((vector_size(16)));

#define N_B   64
#define C_IN  512
#define S_SP  784
#define C_OUT 1024
#define NCS   ((size_t)N_B * C_IN * S_SP)   // floats per [n][c][s] buffer

#define AS1 __attribute__((address_space(1)))
#define AS3 __attribute__((address_space(3)))

// ---------------------------------------------------------------------------
// Async global->LDS copies (CDNA5 GLOBAL_LOAD_ASYNC_TO_LDS_*), with sync
// fallback so the file compiles on toolchains lacking the builtins.
// Probed prototypes (from round-2 diagnostics):
//   _b128(global i32x4*, lds i32x4*, imm offset, imm cpol)
//   _b32 (global int*,   lds int*,   imm offset, imm cpol)
// ---------------------------------------------------------------------------
#if defined(__has_builtin)
#  if __has_builtin(__builtin_amdgcn_global_load_async_to_lds_b128) && \
      __has_builtin(__builtin_amdgcn_global_load_async_to_lds_b32)
#    define HAVE_ASYNC_LDS 1
#  endif
#endif
#ifndef HAVE_ASYNC_LDS
#  define HAVE_ASYNC_LDS 0
#endif

__device__ __forceinline__ void cp_b128(const float* g, float* l) {
#if HAVE_ASYNC_LDS
    __builtin_amdgcn_global_load_async_to_lds_b128((AS1 i32x4*)g, (AS3 i32x4*)l, 0, 0);
#else
    *reinterpret_cast<float4*>(l) = *reinterpret_cast<const float4*>(g);
#endif
}
__device__ __forceinline__ void cp_b32(const float* g, float* l) {
#if HAVE_ASYNC_LDS
    __builtin_amdgcn_global_load_async_to_lds_b32((AS1 int*)g, (AS3 int*)l, 0, 0);
#else
    *l = *g;
#endif
}
__device__ __forceinline__ void cp_wait() {
#if HAVE_ASYNC_LDS
#  if __has_builtin(__builtin_amdgcn_s_wait_asynccnt)
    __builtin_amdgcn_s_wait_asynccnt(0);
#  else
    asm volatile("s_wait_asynccnt 0x0" ::: "memory");
#  endif
#endif
}

// ---------------------------------------------------------------------------
// WMMA helper: D = A(16x4) * B(4x16) + C, fp32 (V_WMMA_F32_16X16X4_F32)
// ---------------------------------------------------------------------------
__device__ __forceinline__ v8f wmma_f32(v2f a, v2f b, v8f c) {
    return __builtin_amdgcn_wmma_f32_16x16x4_f32(
        false, a, false, b, (short)0, c, false, false);
}

// ---------------------------------------------------------------------------
// Channel GEMM over [n][c][s] activations:
//   Y[n][m][s] = sum_c W[m][c] * f(X[n][c][s])   (+ optional residual)
// MODE 0: plain (fully async-staged, double buffered)
// MODE 1: f = relu(GN(.)) prologue (sync B path), += resid epilogue
// Block tile: 256 (m) x 64 (cols), BK = 16. 8 waves, each 32x64 (8 accums).
// ---------------------------------------------------------------------------
template <int MODE>
__global__ __launch_bounds__(256)
void gemm_cnl(const float* __restrict__ W, const float* __restrict__ X,
              float* __restrict__ Y, int Cin, int Mout,
              const float* __restrict__ meanp, const float* __restrict__ rstdp,
              const float* __restrict__ gammap, const float* __restrict__ betap,
              const float* __restrict__ resid)
{
    __shared__ __align__(16) float As[2][256 * 20];  // [row][k] pad 20 (16B-aligned rows)
    __shared__ __align__(16) float Bs[2][16 * 72];   // [k][col] pad 72

    const int tid   = threadIdx.x;
    const int lane  = tid & 31;
    const int wv    = tid >> 5;        // 0..7
    const int laneM = lane & 15;
    const int laneH = lane >> 4;       // 0/1

    const int colBase = blockIdx.x * 64;
    const int mBase   = blockIdx.y * 256;
    const int rBase   = 32 * wv;       // wave's rows within the 256-row tile

    v8f acc[2][4] = {};

    // B-tile thread mapping: 4 contiguous floats each
    const int kRow = tid >> 4;                     // 0..15
    const int cg   = tid & 15;                     // cols cg*4..cg*4+3
    const int col0 = colBase + cg * 4;
    const unsigned nIdx = (unsigned)col0 / 784u;   // 16-col tiles never cross n
    const int s0 = col0 - (int)nIdx * 784;

    auto stageA = [&](int kb, int b) {             // 256x16: one row per thread
        const float* src = W + (size_t)(mBase + tid) * Cin + kb;
        float* dst = &As[b][tid * 20];
        cp_b128(src,      dst);
        cp_b128(src + 4,  dst + 4);
        cp_b128(src + 8,  dst + 8);
        cp_b128(src + 12, dst + 12);
    };
    auto stageB = [&](int kb, int b) {             // 16x64
        const int c = kb + kRow;
        const float* src = X + ((size_t)nIdx * Cin + c) * S_SP + s0;
        float* dst = &Bs[b][kRow * 72 + cg * 4];
        if (MODE == 0) {
            cp_b128(src, dst);
        } else {
            float4 p = *reinterpret_cast<const float4*>(src);
            const float g  = gammap[c] * rstdp[nIdx];
            const float b2 = betap[c] - meanp[nIdx] * g;
            p.x = fmaxf(fmaf(p.x, g, b2), 0.f);
            p.y = fmaxf(fmaf(p.y, g, b2), 0.f);
            p.z = fmaxf(fmaf(p.z, g, b2), 0.f);
            p.w = fmaxf(fmaf(p.w, g, b2), 0.f);
            *reinterpret_cast<float4*>(dst) = p;
        }
    };

    stageA(0, 0);
    stageB(0, 0);
    cp_wait();
    __syncthreads();

    int buf = 0;
    for (int kb = 0; kb < Cin; kb += 16) {
        const int nxt = kb + 16;
        if (nxt < Cin) {                      // prefetch next stage (async)
            stageA(nxt, buf ^ 1);
            stageB(nxt, buf ^ 1);
        }
        #pragma unroll
        for (int kk = 0; kk < 16; kk += 4) {
            v2f a0, a1;
            a0[0] = As[buf][(rBase + laneM) * 20 + kk + 2 * laneH + 0];
            a0[1] = As[buf][(rBase + laneM) * 20 + kk + 2 * laneH + 1];
            a1[0] = As[buf][(rBase + 16 + laneM) * 20 + kk + 2 * laneH + 0];
            a1[1] = As[buf][(rBase + 16 + laneM) * 20 + kk + 2 * laneH + 1];
            #pragma unroll
            for (int ct = 0; ct < 4; ++ct) {
                v2f b;
                b[0] = Bs[buf][(kk + 2 * laneH + 0) * 72 + ct * 16 + laneM];
                b[1] = Bs[buf][(kk + 2 * laneH + 1) * 72 + ct * 16 + laneM];
                acc[0][ct] = wmma_f32(a0, b, acc[0][ct]);
                acc[1][ct] = wmma_f32(a1, b, acc[1][ct]);
            }
        }
        cp_wait();
        __syncthreads();
        buf ^= 1;
    }

    // ---- store ----
    #pragma unroll
    for (int rg = 0; rg < 2; ++rg) {
        #pragma unroll
        for (int ct = 0; ct < 4; ++ct) {
            const int col = colBase + ct * 16 + laneM;
            const unsigned n2 = (unsigned)col / 784u;
            const int s2 = col - (int)n2 * 784;
            #pragma unroll
            for (int r = 0; r < 8; ++r) {
                const int m = mBase + rBase + 16 * rg + r + 8 * laneH;
                const size_t o = ((size_t)n2 * Mout + m) * S_SP + s2;
                float val = acc[rg][ct][r];
                if (MODE == 1) val += resid[o];
                Y[o] = val;
            }
        }
    }
}

// ---------------------------------------------------------------------------
// Batch-axis attention, one block per spatial site s (784 blocks).
//   att = softmax_j( Q_s K_s^T / sqrt(C) )   (64x64, K=512)
//   virt_s = att * V_s                        (64x512, K=64)
// Strided gathers staged via async b32 copies into LDS.
// ---------------------------------------------------------------------------
__global__ __launch_bounds__(256)
void attn_kernel(const float* __restrict__ q, const float* __restrict__ k,
                 const float* __restrict__ v, float* __restrict__ virt)
{
    __shared__ float Qs[64 * 17];     // A-tile [i][kc]
    __shared__ float Ks[16 * 72];     // B-tile [kc][j]
    __shared__ float att[64 * 65];    // attention matrix [i][j]
    __shared__ float Vs[64 * 72];     // B-tile [j][c]

    const int s     = blockIdx.x;
    const int tid   = threadIdx.x;
    const int lane  = tid & 31;
    const int wv    = tid >> 5;
    const int laneM = lane & 15;
    const int laneH = lane >> 4;

    const int i0 = 16 * (wv & 3);     // output row block (batch i)
    const int j0 = 32 * (wv >> 2);    // output col block

    // ---------- phase 1: att = Q K^T ----------
    v8f acc[2] = {};
    for (int kb = 0; kb < C_IN; kb += 16) {
        #pragma unroll
        for (int t = 0; t < 4; ++t) {                  // Qs: 64x16
            const int e = tid + 256 * t;
            const int row = e >> 4, cc = e & 15;
            cp_b32(q + ((size_t)row * C_IN + kb + cc) * S_SP + s,
                   &Qs[row * 17 + cc]);
        }
        #pragma unroll
        for (int t = 0; t < 4; ++t) {                  // Ks: 16x64 (transposed)
            const int e = tid + 256 * t;
            const int kr = e >> 6, j = e & 63;
            cp_b32(k + ((size_t)j * C_IN + kb + kr) * S_SP + s,
                   &Ks[kr * 72 + j]);
        }
        cp_wait();
        __syncthreads();
        #pragma unroll
        for (int kk = 0; kk < 16; kk += 4) {
            v2f a;
            a[0] = Qs[(i0 + laneM) * 17 + kk + 2 * laneH + 0];
            a[1] = Qs[(i0 + laneM) * 17 + kk + 2 * laneH + 1];
            #pragma unroll
            for (int ct = 0; ct < 2; ++ct) {
                v2f b;
                b[0] = Ks[(kk + 2 * laneH + 0) * 72 + j0 + ct * 16 + laneM];
                b[1] = Ks[(kk + 2 * laneH + 1) * 72 + j0 + ct * 16 + laneM];
                acc[ct] = wmma_f32(a, b, acc[ct]);
            }
        }
        __syncthreads();
    }
    const float sc = 0.04419417382415922f;   // 1/sqrt(512)
    #pragma unroll
    for (int ct = 0; ct < 2; ++ct)
        #pragma unroll
        for (int r = 0; r < 8; ++r)
            att[(i0 + r + 8 * laneH) * 65 + j0 + ct * 16 + laneM] = acc[ct][r] * sc;
    __syncthreads();

    // ---------- softmax over j (one row per thread 0..63) ----------
    if (tid < 64) {
        float mx = -3.4e38f;
        for (int j = 0; j < 64; ++j) mx = fmaxf(mx, att[tid * 65 + j]);
        float sum = 0.f;
        for (int j = 0; j < 64; ++j) {
            const float e = __expf(att[tid * 65 + j] - mx);
            att[tid * 65 + j] = e;
            sum += e;
        }
        const float inv = 1.0f / sum;
        for (int j = 0; j < 64; ++j) att[tid * 65 + j] *= inv;
    }
    __syncthreads();

    // ---------- phase 2: virt = att * V ----------
    for (int cb = 0; cb < C_IN; cb += 64) {
        v8f vac[2] = {};
        #pragma unroll
        for (int t = 0; t < 16; ++t) {                 // Vs: 64x64
            const int e = tid + 256 * t;
            const int j = e >> 6, cc = e & 63;
            cp_b32(v + ((size_t)j * C_IN + cb + cc) * S_SP + s,
                   &Vs[j * 72 + cc]);
        }
        cp_wait();
        __syncthreads();
        #pragma unroll
        for (int kk = 0; kk < 64; kk += 4) {
            v2f a;
            a[0] = att[(i0 + laneM) * 65 + kk + 2 * laneH + 0];
            a[1] = att[(i0 + laneM) * 65 + kk + 2 * laneH + 1];
            #pragma unroll
            for (int ct = 0; ct < 2; ++ct) {
                v2f b;
                b[0] = Vs[(kk + 2 * laneH + 0) * 72 + j0 + ct * 16 + laneM];
                b[1] = Vs[(kk + 2 * laneH + 1) * 72 + j0 + ct * 16 + laneM];
                vac[ct] = wmma_f32(a, b, vac[ct]);
            }
        }
        #pragma unroll
        for (int ct = 0; ct < 2; ++ct) {
            const int c = cb + j0 + ct * 16 + laneM;
            #pragma unroll
            for (int r = 0; r < 8; ++r) {
                const int i = i0 + r + 8 * laneH;
                virt[((size_t)i * C_IN + c) * S_SP + s] = vac[ct][r];
            }
        }
        __syncthreads();
    }
}

// ---------------------------------------------------------------------------
// GroupNorm stats: one block per sample; per-n slab is contiguous (C*S floats)
// ---------------------------------------------------------------------------
__global__ __launch_bounds__(256)
void gn_stats(const float* __restrict__ virt, float* __restrict__ meanp,
              float* __restrict__ rstdp)
{
    __shared__ float rs[256], rq[256];
    const int n = blockIdx.x;
    const size_t base = (size_t)n * (C_IN * S_SP);
    const int tot = C_IN * S_SP;                 // 401408
    float s = 0.f, q2 = 0.f;
    for (int i = threadIdx.x; i < tot; i += 256) {
        const float x = virt[base + i];
        s += x; q2 += x * x;
    }
    rs[threadIdx.x] = s; rq[threadIdx.x] = q2;
    __syncthreads();
    for (int st = 128; st > 0; st >>= 1) {
        if (threadIdx.x < st) {
            rs[threadIdx.x] += rs[threadIdx.x + st];
            rq[threadIdx.x] += rq[threadIdx.x + st];
        }
        __syncthreads();
    }
    if (threadIdx.x == 0) {
        const float m   = rs[0] / (float)tot;
        const float var = rq[0] / (float)tot - m * m;
        meanp[n] = m;
        rstdp[n] = rsqrtf(var + 1e-5f);
    }
}

// ---------------------------------------------------------------------------
extern "C" void kernel_launch(void* const* d_in, const int* in_sizes, int n_in,
                              void* d_out, int out_size, void* d_ws, size_t ws_size,
                              hipStream_t stream) {
    const float* x     = (const float*)d_in[0];
    const float* Wq    = (const float*)d_in[1];
    const float* Wk    = (const float*)d_in[2];
    const float* Wv    = (const float*)d_in[3];
    const float* Wc    = (const float*)d_in[4];
    const float* Wo    = (const float*)d_in[5];
    const float* gamma = (const float*)d_in[6];
    const float* beta  = (const float*)d_in[7];

    float* wsf   = (float*)d_ws;
    float* qb    = wsf;                 // [n][c][s]
    float* kb    = wsf + NCS;
    float* vb    = wsf + 2 * NCS;
    float* virt  = wsf + 3 * NCS;
    float* meanp = wsf + 4 * NCS;       // 64 floats
    float* rstdp = meanp + 64;
    float* yb    = qb;                  // reuse q buffer after attention

    const dim3 blk(256);
    const dim3 grid512(S_SP * N_B / 64, C_IN / 256);    // (784, 2)
    const dim3 grid1024(S_SP * N_B / 64, C_OUT / 256);  // (784, 4)

    // Q / K / V projections
    gemm_cnl<0><<<grid512, blk, 0, stream>>>(Wq, x, qb, C_IN, C_IN,
                                             nullptr, nullptr, nullptr, nullptr, nullptr);
    gemm_cnl<0><<<grid512, blk, 0, stream>>>(Wk, x, kb, C_IN, C_IN,
                                             nullptr, nullptr, nullptr, nullptr, nullptr);
    gemm_cnl<0><<<grid512, blk, 0, stream>>>(Wv, x, vb, C_IN, C_IN,
                                             nullptr, nullptr, nullptr, nullptr, nullptr);
    // batch-axis attention per spatial site
    attn_kernel<<<dim3(S_SP), blk, 0, stream>>>(qb, kb, vb, virt);
    // GroupNorm statistics
    gn_stats<<<dim3(N_B), blk, 0, stream>>>(virt, meanp, rstdp);
    // y = x + Wc * relu(GN(virt))   (fused prologue/epilogue)
    gemm_cnl<1><<<grid512, blk, 0, stream>>>(Wc, virt, yb, C_IN, C_IN,
                                             meanp, rstdp, gamma, beta, x);
    // out = Wo * y
    gemm_cnl<0><<<grid1024, blk, 0, stream>>>(Wo, yb, (float*)d_out, C_IN, C_OUT,
                                              nullptr, nullptr, nullptr, nullptr, nullptr);
}